// CTRGC_33122787787415
// MI455X (gfx1250) — compile-verified
//
#include <hip/hip_runtime.h>

typedef float v2f __attribute__((ext_vector_type(2)));
typedef float v8f __attribute__((ext_vector_type(8)));

#define NB   64
#define CCH  64
#define TT   256
#define VV   25
#define OO   64
#define RR   8
#define TVC  (TT*VV)        // 6400, contiguous (t,v) span per (n,c)
#define TCH  32             // t-chunk per block in gemm kernel
#define VPAD 28             // v padded to 28 (7 k-steps of 4) for GEMM2

// ---------------------------------------------------------------------------
// Kernel 1: xm[n,c,v] = mean over t of x[n,c,t,v].  One block per (n,c).
// Thread j (<200) walks flat offsets tb*25 + j: every wave reads 128
// contiguous bytes -> 100% coalesced over the dominant 105 MB read.
// ---------------------------------------------------------------------------
__global__ void __launch_bounds__(256) mean_kernel(const float* __restrict__ x,
                                                   float* __restrict__ xm) {
  int nc = blockIdx.x;                       // n*64 + c
  const float* p = x + (size_t)nc * TVC;
  int j = threadIdx.x;
  float s = 0.0f;
  if (j < 8 * VV) {                          // j = (t%8)*25 + v
    for (int tb = 0; tb < TT; tb += 8) s += p[tb * VV + j];
  }
  __shared__ float part[8 * VV];
  if (j < 8 * VV) part[j] = s;
  __syncthreads();
  if (j < VV) {
    float tot = 0.0f;
    for (int i = 0; i < 8; i++) tot += part[j + i * VV];
    xm[nc * VV + j] = tot * (1.0f / 256.0f);
  }
}

// ---------------------------------------------------------------------------
// Kernel 2: adj[n,o,u,v] = W4@(tanh(x1m[u]-x2m[v]) + tanh(softmax(5*q[u]*k[v])))
//           + 2*b4[o] + A[u,v].   One block per n (tiny work).
// ---------------------------------------------------------------------------
__global__ void __launch_bounds__(256) adj_kernel(
    const float* __restrict__ xm, const float* __restrict__ A,
    const float* __restrict__ W1, const float* __restrict__ b1,
    const float* __restrict__ W2, const float* __restrict__ b2,
    const float* __restrict__ W4, const float* __restrict__ b4,
    const float* __restrict__ W5, const float* __restrict__ b5,
    const float* __restrict__ w6, const float* __restrict__ b6,
    const float* __restrict__ w7, const float* __restrict__ b7,
    float* __restrict__ adj) {
  int n = blockIdx.x;
  int tid = threadIdx.x;
  __shared__ float sxm[CCH * VV];            // 6.4 KB
  __shared__ float s1[RR * VV], s2[RR * VV], sm[RR * VV];
  __shared__ float sS[RR][VV * VV];          // d + res, 20 KB

  for (int i = tid; i < CCH * VV; i += 256) sxm[i] = xm[n * CCH * VV + i];
  __syncthreads();

  // x1m / x2m / m : 3 x [8 x 25] projections of the T-mean
  for (int i = tid; i < 3 * RR * VV; i += 256) {
    int which = i / (RR * VV);
    int rv = i % (RR * VV);
    int r = rv / VV, v = rv % VV;
    const float* W = (which == 0) ? W1 : (which == 1) ? W2 : W5;
    float acc = 0.0f;
    for (int c = 0; c < CCH; c++) acc += W[r * CCH + c] * sxm[c * VV + v];
    float bb = (which == 0) ? b1[r] : (which == 1) ? b2[r] : b5[r];
    float* dst = (which == 0) ? s1 : (which == 1) ? s2 : sm;
    dst[rv] = acc + bb;
  }
  __syncthreads();

  float W6 = w6[0], B6 = b6[0], W7 = w7[0], B7 = b7[0];
  // per (r,u): softmax over v, res = tanh(softmax), d = tanh(x1m[u]-x2m[v])
  for (int ru = tid; ru < RR * VV; ru += 256) {
    int r = ru / VV, u = ru % VV;
    float q = W6 * sm[r * VV + u] + B6;
    float att[VV];
    float mx = -3.0e38f;
    for (int v = 0; v < VV; v++) {
      float k = W7 * sm[r * VV + v] + B7;
      att[v] = 5.0f * q * k;
      mx = fmaxf(mx, att[v]);
    }
    float sum = 0.0f;
    for (int v = 0; v < VV; v++) { att[v] = __expf(att[v] - mx); sum += att[v]; }
    float inv = 1.0f / sum;
    float x1u = s1[r * VV + u];
    for (int v = 0; v < VV; v++) {
      float res = tanhf(att[v] * inv);
      float d   = tanhf(x1u - s2[r * VV + v]);
      sS[r][u * VV + v] = d + res;
    }
  }
  __syncthreads();

  // adj = W4 @ sS + 2*b4 + A
  for (int i = tid; i < OO * VV * VV; i += 256) {
    int o = i / (VV * VV);
    int uv = i % (VV * VV);
    float acc = 2.0f * b4[o] + A[uv];
    for (int r = 0; r < RR; r++) acc += W4[o * RR + r] * sS[r][uv];
    adj[(size_t)(n * OO + o) * (VV * VV) + uv] = acc;
  }
}

// ---------------------------------------------------------------------------
// Kernel 3: fused  x3 = W3@x + b3  (f32 WMMA, K=64)  -> LDS  ->
//           out[t,u] = sum_v x3[t,v] * adj[u,v]      (f32 WMMA, K=28-padded)
// Block = (t-chunk of 32, o-chunk of 16, n).  256 threads = 8 waves.
// ---------------------------------------------------------------------------
__global__ void __launch_bounds__(256) gemm_kernel(
    const float* __restrict__ x,  const float* __restrict__ W3,
    const float* __restrict__ b3, const float* __restrict__ adj,
    float* __restrict__ out) {
  __shared__ float x3s[16 * TCH * VPAD];     // [o_local][t][vpad] = 56 KB

  int tchunk = blockIdx.x;                   // 0..7
  int o0 = blockIdx.y * 16;                  // 0,16,32,48
  int n = blockIdx.z;
  int tid = threadIdx.x;
  int lane = tid & 31;
  int w = tid >> 5;

  int nl = lane & 15;                        // column index within tile
  int khalf = lane >> 4;                     // 0: K+0/1, 1: K+2/3
  int ksel = khalf * 2;
  int mbase = khalf * 8;                     // D-frag rows: mbase..mbase+7

  // zero LDS so v=25..27 padding columns stay zero
  for (int i = tid; i < 16 * TCH * VPAD; i += 256) x3s[i] = 0.0f;
  __syncthreads();

  const float* xn = x + (size_t)n * CCH * TVC + (size_t)tchunk * TCH * VV;

  // --- A fragments of W3 (rows o0..o0+15, all 16 K-steps) as b64 loads ---
  v2f a[16];
  {
    int m = o0 + nl;
    for (int kk = 0; kk < 16; kk++) {
      int k = kk * 4 + ksel;                 // even -> 8B aligned pair
      a[kk] = *(const v2f*)(W3 + m * CCH + k);
    }
  }
  float bias[8];
  for (int r = 0; r < 8; r++) bias[r] = b3[o0 + mbase + r];

  // --- Phase 1: x3 tile = W3 @ x, 50 column tiles of 16 over 800 (t,v) cols ---
  for (int jt = w; jt < 50; jt += 8) {
    int col0 = jt * 16;
    v8f acc = {};
    for (int kk = 0; kk < 16; kk++) {
      int krow = kk * 4 + ksel;
      v2f b;
      b.x = xn[(size_t)krow * TVC + col0 + nl];
      b.y = xn[(size_t)(krow + 1) * TVC + col0 + nl];
      acc = __builtin_amdgcn_wmma_f32_16x16x4_f32(
          false, a[kk], false, b, (short)0, acc, false, false);
    }
    // scatter D tile into padded LDS: col -> (t, v)
    int colA = col0 + nl;
    int tl = colA / VV, v = colA % VV;
    float* dst = &x3s[tl * VPAD + v];
    for (int r = 0; r < 8; r++)
      dst[(mbase + r) * (TCH * VPAD)] = acc[r] + bias[r];
  }
  __syncthreads();

  // --- Phase 2: out[t,u] = sum_v x3[t,v] * adj[o][u][v] ---
  // 16 o x 2 M-tiles(t) x 2 N-tiles(u) = 64 assignments over 8 waves
  for (int idx = w; idx < 64; idx += 8) {
    int o_l = idx >> 2;
    int Mt = (idx >> 1) & 1;
    int Nt = idx & 1;
    const float* adjn = adj + (size_t)(n * OO + o0 + o_l) * (VV * VV);
    const float* xrow = &x3s[o_l * (TCH * VPAD)];
    int trow = Mt * 16 + nl;
    int u = Nt * 16 + nl;
    v8f acc = {};
    for (int k0 = 0; k0 < VPAD; k0 += 4) {
      int v = k0 + ksel;                     // even -> 8B aligned LDS pair
      v2f a2 = *(const v2f*)(xrow + trow * VPAD + v);
      v2f b2;
      b2.x = (u < VV && v < VV)     ? adjn[u * VV + v]     : 0.0f;
      b2.y = (u < VV && v + 1 < VV) ? adjn[u * VV + v + 1] : 0.0f;
      acc = __builtin_amdgcn_wmma_f32_16x16x4_f32(
          false, a2, false, b2, (short)0, acc, false, false);
    }
    if (u < VV) {
      size_t t = (size_t)tchunk * TCH + Mt * 16 + mbase;
      float* op = out + ((size_t)(n * OO + o0 + o_l) * TT + t) * VV + u;
      for (int r = 0; r < 8; r++) op[r * VV] = acc[r];
    }
  }
}

// ---------------------------------------------------------------------------
extern "C" void kernel_launch(void* const* d_in, const int* in_sizes, int n_in,
                              void* d_out, int out_size, void* d_ws, size_t ws_size,
                              hipStream_t stream) {
  const float* x  = (const float*)d_in[0];
  const float* A  = (const float*)d_in[1];
  const float* W1 = (const float*)d_in[2];
  const float* b1 = (const float*)d_in[3];
  const float* W2 = (const float*)d_in[4];
  const float* b2 = (const float*)d_in[5];
  const float* W3 = (const float*)d_in[6];
  const float* b3 = (const float*)d_in[7];
  const float* W4 = (const float*)d_in[8];
  const float* b4 = (const float*)d_in[9];
  const float* W5 = (const float*)d_in[10];
  const float* b5 = (const float*)d_in[11];
  const float* w6 = (const float*)d_in[12];
  const float* b6 = (const float*)d_in[13];
  const float* w7 = (const float*)d_in[14];
  const float* b7 = (const float*)d_in[15];
  float* outp = (float*)d_out;

  float* xm   = (float*)d_ws;                      // [64*64*25]
  float* adjw = xm + NB * CCH * VV;                // [64*64*25*25]

  mean_kernel<<<NB * CCH, 256, 0, stream>>>(x, xm);
  adj_kernel<<<NB, 256, 0, stream>>>(xm, A, W1, b1, W2, b2, W4, b4, W5, b5,
                                     w6, b6, w7, b7, adjw);
  gemm_kernel<<<dim3(TT / TCH, OO / 16, NB), 256, 0, stream>>>(
      x, W3, b3, adjw, outp);
}